// GCNLayer_73126113181909
// MI455X (gfx1250) — compile-verified
//
#include <hip/hip_runtime.h>

#define D 128
#define LDA 132          // A-tile LDS row stride (floats): 4m mod 64 spread -> conflict-free
#define LDP 144          // B pair-row stride in float2 (288 words == 32 mod 64 -> half-wave split)
#define NKP (D / 2)      // 64 k-pair rows

typedef float v2f __attribute__((ext_vector_type(2)));
typedef float v8f __attribute__((ext_vector_type(8)));

// ---------------------------------------------------------------------------
// Phase 1: row-parallel SpMM. edge_row is sorted -> each row owns a contiguous
// edge segment. One wave per row, no atomics, deterministic.
// Edge (col,val) pairs are loaded coalesced 32-at-a-time and broadcast with
// v_readlane, so the steady-state loop issues ~1 VMEM request per edge
// (the 512B x-row gather) instead of 3.
// ---------------------------------------------------------------------------
__device__ __forceinline__ int lower_bound_i32(const int* __restrict__ a, int n, int key) {
    int lo = 0, hi = n;
    while (lo < hi) {
        int mid = (lo + hi) >> 1;
        if (a[mid] < key) lo = mid + 1; else hi = mid;
    }
    return lo;
}

__global__ __launch_bounds__(256) void gcn_spmm_rows(
    const float* __restrict__ x,
    const int*   __restrict__ erow,
    const int*   __restrict__ ecol,
    const float* __restrict__ eval_,
    float*       __restrict__ h,
    int nEdges, int nRows)
{
    const int wavesPerBlock = blockDim.x >> 5;
    const int wave = threadIdx.x >> 5;
    const int lane = threadIdx.x & 31;
    const int row  = blockIdx.x * wavesPerBlock + wave;
    if (row >= nRows) return;

    // Segment bounds are wave-uniform: pin them to SGPRs.
    const int lo = __builtin_amdgcn_readfirstlane(lower_bound_i32(erow, nEdges, row));
    const int hi = __builtin_amdgcn_readfirstlane(lower_bound_i32(erow, nEdges, row + 1));

    float4 acc = make_float4(0.f, 0.f, 0.f, 0.f);
    const int fo = lane * 4;                    // this lane's 4 features

    for (int base = lo; base < hi; base += 32) {
        const int idx = base + lane;
        int   cL = 0;
        float vL = 0.f;
        if (idx < hi) {                         // coalesced edge-chunk load
            cL = ecol[idx];
            vL = eval_[idx];
        }
        const int n = hi - base;                // scalar
        #pragma unroll
        for (int j = 0; j < 32; ++j) {
            if (j < n) {                        // scalar compare+branch
                const int   c = __builtin_amdgcn_readlane(cL, j);
                const float v = __int_as_float(
                    __builtin_amdgcn_readlane(__float_as_int(vL), j));
                const float4 xs =
                    *reinterpret_cast<const float4*>(x + (size_t)c * D + fo);
                acc.x = fmaf(v, xs.x, acc.x);
                acc.y = fmaf(v, xs.y, acc.y);
                acc.z = fmaf(v, xs.z, acc.z);
                acc.w = fmaf(v, xs.w, acc.w);
            }
        }
    }
    *reinterpret_cast<float4*>(h + (size_t)row * D + fo) = acc;
}

// ---------------------------------------------------------------------------
// Phase 2: out = h @ W with V_WMMA_F32_16X16X4_F32, persistent row-tile loop.
// W is staged ONCE per block in LDS in B-operand pair layout:
//   ldsB[p * 2*LDP + 2n] = { W[2p][n], W[2p+1][n] }      (p = k/2)
// so each WMMA step's B operand is a single conflict-free ds_load_b64.
//
// Operand layouts (ISA 7.12.2, wave32):
//   A 16x4 : lane m=lane&15, kh=lane>>4; VGPR0=K(2kh), VGPR1=K(2kh+1)
//   B 4x16 : lane n=lane&15, kh=lane>>4; VGPR0=row(2kh), VGPR1=row(2kh+1)
//   C/D    : VGPR v -> row v + 8*(lane>>4), col = lane&15
// ---------------------------------------------------------------------------
__global__ __launch_bounds__(256) void gcn_gemm_wmma(
    const float* __restrict__ h,
    const float* __restrict__ W,
    float*       __restrict__ out,
    int nRows, int nTiles)
{
    __shared__ float ldsA[16 * LDA];            //  8.4 KB
    __shared__ float ldsB[NKP * 2 * LDP];       // 73.7 KB

    const int tid  = threadIdx.x;
    const int wave = tid >> 5;
    const int lane = tid & 31;

    // ---- one-time W staging in pair layout (coalesced b32 reads) ----
    for (int i = tid; i < NKP * D; i += 256) {  // 8192 float2 pairs
        const int p = i >> 7;                   // k-pair row
        const int n = i & (D - 1);              // column
        const int k = p << 1;
        float2 f2;
        f2.x = W[(size_t)k       * D + n];
        f2.y = W[(size_t)(k + 1) * D + n];
        *reinterpret_cast<float2*>(&ldsB[p * (2 * LDP) + 2 * n]) = f2;
    }

    const int col0 = wave * 16;
    const int m    = lane & 15;                 // A row / B,D column
    const int kh   = lane >> 4;

    for (int tile = blockIdx.x; tile < nTiles; tile += gridDim.x) {
        const int row0 = tile * 16;

        __syncthreads();                        // ldsA safe to overwrite
        for (int i = tid; i < 16 * (D / 4); i += 256) {
            const int r  = i >> 5;
            const int c4 = i & 31;
            const int rg = row0 + r;
            float4 v = make_float4(0.f, 0.f, 0.f, 0.f);
            if (rg < nRows)
                v = *reinterpret_cast<const float4*>(h + (size_t)rg * D + c4 * 4);
            *reinterpret_cast<float4*>(&ldsA[r * LDA + c4 * 4]) = v;
        }
        __syncthreads();                        // ldsA (and first-pass ldsB) ready

        v8f c = {};
        #pragma unroll
        for (int k0 = 0; k0 < D; k0 += 4) {
            const int k = k0 + 2 * kh;
            v2f a, b;
            a.x = ldsA[m * LDA + k];
            a.y = ldsA[m * LDA + k + 1];
            b   = *reinterpret_cast<const v2f*>(
                      &ldsB[(k >> 1) * (2 * LDP) + 2 * (col0 + m)]);
            c = __builtin_amdgcn_wmma_f32_16x16x4_f32(
                    /*neg_a=*/false, a, /*neg_b=*/false, b,
                    /*c_mod=*/(short)0, c, /*reuse_a=*/false, /*reuse_b=*/false);
        }

        #pragma unroll
        for (int v = 0; v < 8; ++v) {
            const int rg = row0 + v + 8 * kh;
            if (rg < nRows)
                out[(size_t)rg * D + col0 + m] = c[v];
        }
    }
}

// ---------------------------------------------------------------------------
extern "C" void kernel_launch(void* const* d_in, const int* in_sizes, int n_in,
                              void* d_out, int out_size, void* d_ws, size_t ws_size,
                              hipStream_t stream)
{
    const float* x     = (const float*)d_in[0];
    const int*   erow  = (const int*)  d_in[1];
    const int*   ecol  = (const int*)  d_in[2];
    const float* eval_ = (const float*)d_in[3];
    const float* W     = (const float*)d_in[4];
    float*       out   = (float*)d_out;
    float*       h     = (float*)d_ws;          // nRows * 128 floats (~51.2 MB)

    const int nEdges = in_sizes[1];
    const int nRows  = in_sizes[0] / D;

    // Phase 1: one wave per row, 8 waves per 256-thread block.
    const int wavesPerBlock = 8;
    const int spmmBlocks = (nRows + wavesPerBlock - 1) / wavesPerBlock;
    gcn_spmm_rows<<<spmmBlocks, 256, 0, stream>>>(x, erow, ecol, eval_, h, nEdges, nRows);

    // Phase 2: persistent blocks striding over 16-row tiles.
    const int nTiles = (nRows + 15) / 16;
    const int gemmBlocks = nTiles < 1024 ? nTiles : 1024;
    gcn_gemm_wmma<<<gemmBlocks, 256, 0, stream>>>(h, W, out, nRows, nTiles);
}